// BatchAllTripletLoss_50491635532230
// MI455X (gfx1250) — compile-verified
//
#include <hip/hip_runtime.h>

#define N 512
#define DIM 256
#define MARGIN 5.0f
#define EPSF 1e-16f

typedef __attribute__((ext_vector_type(2))) float v2f;
typedef __attribute__((ext_vector_type(8))) float v8f;

// ---------------------------------------------------------------------------
// Kernel 1: row squared-norms sq[i] = sum_k rep[i,k]^2   (512 rows x 256)
// ---------------------------------------------------------------------------
__global__ void rownorm_kernel(const float* __restrict__ rep,
                               float* __restrict__ sq) {
    int row = blockIdx.x * blockDim.x + threadIdx.x;
    if (row >= N) return;
    const float4* r = (const float4*)(rep + row * DIM);
    float s = 0.0f;
#pragma unroll 8
    for (int i = 0; i < DIM / 4; ++i) {
        float4 v = r[i];
        s += v.x * v.x + v.y * v.y + v.z * v.z + v.w * v.w;
    }
    sq[row] = s;
}

// ---------------------------------------------------------------------------
// Kernel 2: Gram via V_WMMA_F32_16X16X4_F32, then fused distance transform.
// One wave per 16x16 tile of G = R * R^T. A-layout (32-bit 16x4): lane L holds
// row M=L%16, K-pair at 2*(L/16); B-layout mirrors with N=L%16. So both
// operands are a single float2 load per WMMA step.
// ---------------------------------------------------------------------------
__global__ void gram_dist_kernel(const float* __restrict__ rep,
                                 const float* __restrict__ sq,
                                 float* __restrict__ pd) {
    const int wave = (blockIdx.x * blockDim.x + threadIdx.x) >> 5;
    const int lane = threadIdx.x & 31;
    const int tI   = (wave >> 5) << 4;   // tile row base (wave/32 * 16)
    const int tJ   = (wave & 31) << 4;   // tile col base (wave%32 * 16)
    const int half = lane >> 4;          // 0: lanes 0-15, 1: lanes 16-31
    const int lm   = lane & 15;

    const float* rowA = rep + (tI + lm) * DIM + half * 2;
    const float* rowB = rep + (tJ + lm) * DIM + half * 2;

    v8f acc = {};
#pragma unroll 4
    for (int k0 = 0; k0 < DIM; k0 += 4) {
        v2f a = *(const v2f*)(rowA + k0);
        v2f b = *(const v2f*)(rowB + k0);
        // 8 args: (neg_a, A, neg_b, B, c_mod, C, reuse_a, reuse_b)
        acc = __builtin_amdgcn_wmma_f32_16x16x4_f32(
            false, a, false, b, (short)0, acc, false, false);
    }

    // C/D layout: lane holds column N = lm; VGPR v holds row M = v + 8*half.
    const int col  = tJ + lm;
    const float sqj = sq[col];
#pragma unroll
    for (int v = 0; v < 8; ++v) {
        const int row = tI + v + half * 8;
        float d2 = sq[row] - 2.0f * acc[v] + sqj;
        d2 = fmaxf(d2, 0.0f);
        pd[row * N + col] = (d2 > 0.0f) ? sqrtf(d2) : 0.0f;
    }
}

// ---------------------------------------------------------------------------
// Kernel 3: per-anchor triplet reduction. One block per anchor a; the anchor's
// distance row + labels live in LDS. (a,p) validity is block-uniform, so the
// p-loop skips invalid positives without divergence; threads split the k axis.
// Deterministic: per-block partials, no float atomics.
// ---------------------------------------------------------------------------
__global__ void triplet_kernel(const float* __restrict__ pd,
                               const int* __restrict__ labels,
                               float* __restrict__ psum,
                               float* __restrict__ pcnt) {
    __shared__ float drow[N];
    __shared__ int   slab[N];
    __shared__ float rs[256];
    __shared__ float rc[256];

    const int a   = blockIdx.x;
    const int tid = threadIdx.x;

    for (int i = tid; i < N; i += blockDim.x) {
        drow[i] = pd[a * N + i];
        slab[i] = labels[i];
    }
    __syncthreads();

    const int la = slab[a];
    float sum = 0.0f, cnt = 0.0f;

    for (int p = 0; p < N; ++p) {
        if (p == a || slab[p] != la) continue;         // uniform branch
        const float dpp = drow[p] + MARGIN;
        for (int k = tid; k < N; k += blockDim.x) {
            if (slab[k] == la) continue;               // implies k!=a, k!=p
            const float v = dpp - drow[k];
            if (v > EPSF) { sum += v; cnt += 1.0f; }
            else if (v > 0.0f) { sum += v; }           // clamp keeps v, not counted
        }
    }

    rs[tid] = sum;
    rc[tid] = cnt;
    __syncthreads();
    for (int s = 128; s > 0; s >>= 1) {
        if (tid < s) { rs[tid] += rs[tid + s]; rc[tid] += rc[tid + s]; }
        __syncthreads();
    }
    if (tid == 0) { psum[a] = rs[0]; pcnt[a] = rc[0]; }
}

// ---------------------------------------------------------------------------
// Kernel 4: final reduce of 512 partials -> scalar loss.
// ---------------------------------------------------------------------------
__global__ void finalize_kernel(const float* __restrict__ psum,
                                const float* __restrict__ pcnt,
                                float* __restrict__ out) {
    __shared__ float rs[256];
    __shared__ float rc[256];
    const int tid = threadIdx.x;
    rs[tid] = psum[tid] + psum[tid + 256];
    rc[tid] = pcnt[tid] + pcnt[tid + 256];
    __syncthreads();
    for (int s = 128; s > 0; s >>= 1) {
        if (tid < s) { rs[tid] += rs[tid + s]; rc[tid] += rc[tid + s]; }
        __syncthreads();
    }
    if (tid == 0) out[0] = rs[0] / (rc[0] + EPSF);
}

extern "C" void kernel_launch(void* const* d_in, const int* in_sizes, int n_in,
                              void* d_out, int out_size, void* d_ws, size_t ws_size,
                              hipStream_t stream) {
    (void)in_sizes; (void)n_in; (void)out_size; (void)ws_size;
    const float* rep    = (const float*)d_in[0];   // [512, 256] f32
    const int*   labels = (const int*)d_in[1];     // [512] i32
    float* out = (float*)d_out;                    // scalar f32

    float* ws   = (float*)d_ws;
    float* sq   = ws;              // 512
    float* psum = ws + 512;        // 512
    float* pcnt = ws + 1024;       // 512
    float* pd   = ws + 1536;       // 512*512

    rownorm_kernel<<<2, 256, 0, stream>>>(rep, sq);
    // 1024 tiles of 16x16, 8 waves (one tile each) per 256-thread block
    gram_dist_kernel<<<128, 256, 0, stream>>>(rep, sq, pd);
    triplet_kernel<<<N, 256, 0, stream>>>(pd, labels, psum, pcnt);
    finalize_kernel<<<1, 256, 0, stream>>>(psum, pcnt, out);
}